// FlexDecodingModule_13073880449551
// MI455X (gfx1250) — compile-verified
//
#include <hip/hip_runtime.h>
#include <hip/hip_bf16.h>
#include <math.h>

// ---------------------------------------------------------------------------
// Problem constants (match reference)
// ---------------------------------------------------------------------------
#define DIMX      1024
#define HEADS     16
#define HEAD_DIM  64
#define MAX_SEQ   4096
#define BATCH     16

typedef __attribute__((ext_vector_type(2))) float v2f;
typedef __attribute__((ext_vector_type(8))) float v8f;

// ---------------------------------------------------------------------------
// Kernel 1/3: fp32 GEMM via V_WMMA_F32_16X16X4_F32.
//   out[b, n] = sum_k act[b, k] * W[n, k]        (act: [16,1024], W: [1024,1024])
// One wave (32 threads) per 16-wide output tile; M = 16 = batch fills the tile.
//
// A (16x4 f32) layout: lanes 0-15 -> M=lane, K-pair {0,1}; lanes 16-31 -> M=lane-16,
//   K-pair {2,3}; VGPR0 = first K of pair, VGPR1 = second.
// B (4x16 f32) layout (mirror of A): lane&15 = N, lane half selects K-pair.
// C/D (16x16 f32): VGPR j -> M = j + 8*(lane>=16), N = lane&15.
// ---------------------------------------------------------------------------
__global__ __launch_bounds__(32)
void proj_wmma_f32(const float* __restrict__ act,   // [16, 1024]
                   const float* __restrict__ W,     // [1024, 1024] row-major W[n][k]
                   float* __restrict__ out)         // [16, 1024]
{
    const int lane  = threadIdx.x;          // 0..31
    const int tileN = blockIdx.x * 16;      // output column base
    const int mn    = lane & 15;            // M for A, N for B
    const int kh    = (lane >> 4) * 2;      // K sub-offset within step: 0 or 2

    const float* arow = act + (size_t)mn * DIMX;                 // batch row (A)
    const float* wrow = W   + (size_t)(tileN + mn) * DIMX;       // weight row (B col)

    v8f acc = {};
    for (int k0 = 0; k0 < DIMX; k0 += 4) {
        v2f a = *(const v2f*)(arow + k0 + kh);   // token[m][k0+kh], [k0+kh+1]
        v2f b = *(const v2f*)(wrow + k0 + kh);   // W[n][k0+kh],     [k0+kh+1]
        acc = __builtin_amdgcn_wmma_f32_16x16x4_f32(
                  /*neg_a=*/false, a, /*neg_b=*/false, b,
                  /*c_mod=*/(short)0, acc, /*reuse_a=*/false, /*reuse_b=*/false);
    }

    const int mBase = (lane >> 4) * 8;
    const int n     = tileN + mn;
#pragma unroll
    for (int j = 0; j < 8; ++j) {
        out[(size_t)(mBase + j) * DIMX + n] = acc[j];
    }
}

// ---------------------------------------------------------------------------
// Kernel 2: streaming attention for one (batch, head) per block.
// 256 threads: tid = r*16 + c. Row r walks sequence positions (stride 16),
// column c owns head dims 4c..4c+3 (float4). Each 16-lane half-wave loads one
// contiguous 256B K/V row -> fully coalesced; K/V each read exactly once.
// ---------------------------------------------------------------------------
__global__ __launch_bounds__(256)
void attn_decode(const float* __restrict__ kc,     // [B, MAX_SEQ, H, 64]
                 const float* __restrict__ vc,     // [B, MAX_SEQ, H, 64]
                 const float* __restrict__ qbuf,   // [B, 1024]
                 const float* __restrict__ knew,   // [B, 1024]
                 const float* __restrict__ vnew,   // [B, 1024]
                 float* __restrict__ obuf,         // [B, 1024]
                 const int* __restrict__ posPtr)
{
    const int b   = blockIdx.x;     // 0..15
    const int h   = blockIdx.y;     // 0..15
    const int tid = threadIdx.x;    // 0..255
    const int c   = tid & 15;       // d-group: d = 4c..4c+3
    const int r   = tid >> 4;       // 0..15 position row

    const int position = *posPtr;   // 4095
    const int S        = position + 1;
    const float scale  = 0.125f;    // 1/sqrt(64)

    __shared__ float sc[MAX_SEQ];        // 16 KB scores
    __shared__ float red[256];           // block reductions
    __shared__ float outp[16][HEAD_DIM]; // partial outputs per row

    // q for this (b,h): lane's 4 components
    const float4 q4 = ((const float4*)(qbuf + (size_t)b * DIMX + h * HEAD_DIM))[c];

    const size_t bhOff   = ((size_t)b * MAX_SEQ * HEADS + h) * HEAD_DIM;
    const size_t sStride = (size_t)HEADS * HEAD_DIM;            // 1024 floats
    const float* knewRow = knew + (size_t)b * DIMX + h * HEAD_DIM;
    const float* vnewRow = vnew + (size_t)b * DIMX + h * HEAD_DIM;

    // ---- pass 1: scores = (q . k) * scale ----
    for (int s = r; s < S; s += 16) {
        const float* krow = (s == position) ? knewRow : (kc + bhOff + (size_t)s * sStride);
        const float4 k4 = ((const float4*)krow)[c];
        float d = q4.x * k4.x + q4.y * k4.y + q4.z * k4.z + q4.w * k4.w;
        d += __shfl_xor(d, 8, 32);
        d += __shfl_xor(d, 4, 32);
        d += __shfl_xor(d, 2, 32);
        d += __shfl_xor(d, 1, 32);
        if (c == 0) sc[s] = d * scale;
    }
    __syncthreads();

    // ---- softmax: max ----
    float lm = -INFINITY;
    for (int s = tid; s < S; s += 256) lm = fmaxf(lm, sc[s]);
    red[tid] = lm;
    __syncthreads();
    for (int off = 128; off > 0; off >>= 1) {
        if (tid < off) red[tid] = fmaxf(red[tid], red[tid + off]);
        __syncthreads();
    }
    const float m = red[0];
    __syncthreads();

    // ---- softmax: exp + sum ----
    float ls = 0.0f;
    for (int s = tid; s < S; s += 256) {
        const float p = __expf(sc[s] - m);
        sc[s] = p;
        ls += p;
    }
    red[tid] = ls;
    __syncthreads();
    for (int off = 128; off > 0; off >>= 1) {
        if (tid < off) red[tid] += red[tid + off];
        __syncthreads();
    }
    const float inv = 1.0f / red[0];
    __syncthreads();

    // ---- pass 2: out = sum_s p[s] * V[s] ----
    float4 acc = make_float4(0.f, 0.f, 0.f, 0.f);
    for (int s = r; s < S; s += 16) {
        const float* vrow = (s == position) ? vnewRow : (vc + bhOff + (size_t)s * sStride);
        const float4 v4 = ((const float4*)vrow)[c];
        const float p = sc[s];
        acc.x += p * v4.x; acc.y += p * v4.y; acc.z += p * v4.z; acc.w += p * v4.w;
    }
    ((float4*)&outp[r][0])[c] = acc;
    __syncthreads();

    // reduce 16 row-partials; threads 0..63 each own one head dim
    if (tid < HEAD_DIM) {
        float sum = 0.0f;
#pragma unroll
        for (int rr = 0; rr < 16; ++rr) sum += outp[rr][tid];
        obuf[(size_t)b * DIMX + h * HEAD_DIM + tid] = sum * inv;
    }
}

// ---------------------------------------------------------------------------
// kernel_launch
// ---------------------------------------------------------------------------
extern "C" void kernel_launch(void* const* d_in, const int* in_sizes, int n_in,
                              void* d_out, int out_size, void* d_ws, size_t ws_size,
                              hipStream_t stream) {
    (void)in_sizes; (void)n_in; (void)out_size; (void)ws_size;

    const float* token = (const float*)d_in[0];   // [16, 1, 1024]
    const float* kc    = (const float*)d_in[1];   // [16, 4096, 16, 64]
    const float* vc    = (const float*)d_in[2];   // [16, 4096, 16, 64]
    const float* Wq    = (const float*)d_in[3];   // [1024, 1024]
    const float* Wk    = (const float*)d_in[4];
    const float* Wv    = (const float*)d_in[5];
    const float* Wo    = (const float*)d_in[6];
    const int*   pos   = (const int*)d_in[7];     // scalar (4095)

    float* ws   = (float*)d_ws;
    float* qbuf = ws;                      // 16*1024
    float* knew = ws + 1 * BATCH * DIMX;   // 16*1024
    float* vnew = ws + 2 * BATCH * DIMX;   // 16*1024
    float* attn = ws + 3 * BATCH * DIMX;   // 16*1024
    float* outF = (float*)d_out;           // [16, 1, 1024]

    const dim3 projGrid(DIMX / 16);        // 64 tiles of 16 output cols
    const dim3 projBlk(32);                // one wave per tile

    // q / k_new / v_new projections (WMMA f32 16x16x4)
    proj_wmma_f32<<<projGrid, projBlk, 0, stream>>>(token, Wq, qbuf);
    proj_wmma_f32<<<projGrid, projBlk, 0, stream>>>(token, Wk, knew);
    proj_wmma_f32<<<projGrid, projBlk, 0, stream>>>(token, Wv, vnew);

    // attention: one block per (batch, head); streams 512 MB of K/V once
    attn_decode<<<dim3(BATCH, HEADS), dim3(256), 0, stream>>>(
        kc, vc, qbuf, knew, vnew, attn, pos);

    // output projection -> d_out
    proj_wmma_f32<<<projGrid, projBlk, 0, stream>>>(attn, Wo, outF);
}